// AttentionBlockSE3_6433861009744
// MI455X (gfx1250) — compile-verified
//
#include <hip/hip_runtime.h>

// SE(3) graph attention: E=500k edges, N=50k nodes, H=8 heads, 16 feats/head.
// Memory-bound (~560MB min traffic -> ~24us @ 23.3 TB/s). Three passes; keys and
// values streamed exactly once; node-gathered data stays L2-resident (192MB L2).
// Logits computed with V_WMMA_F32_16X16X4_F32 (gram-diagonal trick).

typedef __attribute__((ext_vector_type(2))) float v2f;
typedef __attribute__((ext_vector_type(8))) float v8f;

#define NHEADS 8
#define RSQRT128 0.08838834764831845f /* 1/sqrt(128) */

__device__ __forceinline__ void atomicMaxF(float* addr, float val) {
    int* ai = (int*)addr;
    int old = __float_as_int(*addr);
    while (__int_as_float(old) < val) {
        int assumed = old;
        old = atomicCAS(ai, assumed, __float_as_int(val));
        if (old == assumed) break;
    }
}

// Zero outputs/sums, set running max to -inf.
__global__ void se3_init_kernel(float* __restrict__ out, long long out_n,
                                float* __restrict__ mmax, float* __restrict__ ssum,
                                long long nh) {
    long long i = (long long)blockIdx.x * blockDim.x + threadIdx.x;
    if (i < out_n) out[i] = 0.0f;
    if (i < nh) { mmax[i] = -3.402823466e+38f; ssum[i] = 0.0f; }
}

// Pass A: logits[e,h] = (k[e,h,:16] . q[dst[e],h,:16]) / sqrt(128), atomic-max into m.
// One wave handles 16 (edge,head) pairs via a 16x16 gram tile:
//   D = K_tile(16x16feat) x Q_tile^T, built from 4 chained 16x16x4 f32 WMMAs.
// Diagonal D[i,i] are the wanted dots (ISA C/D layout: comp g -> lane g = D[g,g],
// lane 24+g = D[8+g,8+g]); recovered with one shuffle.
__global__ __launch_bounds__(256)
void se3_logits_wmma_kernel(const float* __restrict__ key0, const float* __restrict__ key1,
                            const float* __restrict__ query0, const float* __restrict__ query1,
                            const int* __restrict__ dst,
                            float* __restrict__ logits, float* __restrict__ mmax,
                            int npairs) {
    const int lane = threadIdx.x & 31;
    const int wave = threadIdx.x >> 5;
    const int tile = blockIdx.x * 8 + wave;
    const int idx  = lane & 15;   // row (pair) index within tile, also B column
    const int hi   = lane >> 4;   // which K half this lane supplies

    int p = tile * 16 + idx;
    if (p >= npairs) p = npairs - 1;      // safety clamp (grid is exact: 4M % 16 == 0)
    const int e = p >> 3;
    const int h = p & 7;
    const int n = dst[e];

    // Per-(e,h) feature vector f[0..15]: f0..3 from degree-0 (C-block h*4..h*4+3),
    // f4..15 from degree-1 ([E,8,12] view: base e*96 + h*12).
    const float* k0 = key0   + (size_t)e * 32 + h * 4;
    const float* k1 = key1   + (size_t)e * 96 + h * 12;
    const float* q0 = query0 + (size_t)n * 32 + h * 4;
    const float* q1 = query1 + (size_t)n * 96 + h * 12;

    __builtin_prefetch(k1 + 3072, 0, 1);  // stream keys ~32 edges ahead

    v8f acc = {};
    // K-chunk 0: features 0..3 (degree-0). A lane holds {A[idx,2hi],A[idx,2hi+1]},
    // B lane holds {B[2hi,idx],B[2hi+1,idx]} = same float2 gather from q features.
    {
        v2f a = *(const v2f*)(k0 + 2 * hi);
        v2f b = *(const v2f*)(q0 + 2 * hi);
        acc = __builtin_amdgcn_wmma_f32_16x16x4_f32(false, a, false, b,
                                                    (short)0, acc, false, false);
    }
    // K-chunks 1..3: features 4..15 (degree-1).
#pragma unroll
    for (int c = 1; c < 4; ++c) {
        const int fo = 4 * (c - 1) + 2 * hi;
        v2f a = *(const v2f*)(k1 + fo);
        v2f b = *(const v2f*)(q1 + fo);
        acc = __builtin_amdgcn_wmma_f32_16x16x4_f32(false, a, false, b,
                                                    (short)0, acc, false, false);
    }

    // Diagonal extraction: component (lane&7) holds a diagonal element at
    // lanes g and 24+g. Lane idx wants D[idx,idx] from lane (idx<8 ? idx : idx+16).
    const int sel = lane & 7;
    float comp = acc[0];
    comp = (sel == 1) ? acc[1] : comp;
    comp = (sel == 2) ? acc[2] : comp;
    comp = (sel == 3) ? acc[3] : comp;
    comp = (sel == 4) ? acc[4] : comp;
    comp = (sel == 5) ? acc[5] : comp;
    comp = (sel == 6) ? acc[6] : comp;
    comp = (sel == 7) ? acc[7] : comp;
    const int src = (idx < 8) ? idx : idx + 16;
    float dot = __shfl(comp, src, 32);

    if (lane < 16) {
        float lg = dot * RSQRT128;
        logits[p] = lg;
        atomicMaxF(&mmax[(size_t)n * NHEADS + h], lg);
    }
}

// Pass B: ex = exp(logit - m[dst]); atomic-add into per-node sums; ex overwrites logits.
__global__ __launch_bounds__(256)
void se3_exp_kernel(float* __restrict__ lg, const int* __restrict__ dst,
                    const float* __restrict__ mmax, float* __restrict__ ssum,
                    int npairs) {
    int p = blockIdx.x * blockDim.x + threadIdx.x;
    if (p >= npairs) return;
    const int e = p >> 3;
    const int h = p & 7;
    const int n = dst[e];
    float ex = expf(lg[p] - mmax[(size_t)n * NHEADS + h]);
    lg[p] = ex;
    atomicAdd(&ssum[(size_t)n * NHEADS + h], ex);
}

// Pass C: w = ex/s; scatter w*value into node outputs. 128 elements per edge
// (32 degree-0 + 96 degree-1), one thread per element; out regions are L2-resident.
__global__ __launch_bounds__(256)
void se3_agg_kernel(const float* __restrict__ value0, const float* __restrict__ value1,
                    const int* __restrict__ dst, const float* __restrict__ ex,
                    const float* __restrict__ ssum,
                    float* __restrict__ out0, float* __restrict__ out1,
                    long long total) {
    long long t = (long long)blockIdx.x * blockDim.x + threadIdx.x;
    if (t >= total) return;
    const int e = (int)(t >> 7);
    const int j = (int)(t & 127);
    const int n = dst[e];

    __builtin_prefetch(value1 + (size_t)e * 96 + 3072, 0, 1);  // stream values ahead

    if (j < 32) {
        const int h = j >> 2;                      // head = channel/4
        float w = ex[(size_t)e * NHEADS + h] / ssum[(size_t)n * NHEADS + h];
        atomicAdd(&out0[(size_t)n * 32 + j], w * value0[(size_t)e * 32 + j]);
    } else {
        const int jj = j - 32;                     // degree-1 element, head = jj/12
        const int h = jj / 12;
        float w = ex[(size_t)e * NHEADS + h] / ssum[(size_t)n * NHEADS + h];
        atomicAdd(&out1[(size_t)n * 96 + jj], w * value1[(size_t)e * 96 + jj]);
    }
}

extern "C" void kernel_launch(void* const* d_in, const int* in_sizes, int n_in,
                              void* d_out, int out_size, void* d_ws, size_t ws_size,
                              hipStream_t stream) {
    const float* key0   = (const float*)d_in[0];
    const float* key1   = (const float*)d_in[1];
    const float* value0 = (const float*)d_in[2];
    const float* value1 = (const float*)d_in[3];
    const float* query0 = (const float*)d_in[4];
    const float* query1 = (const float*)d_in[5];
    const int*   dst    = (const int*)d_in[6];

    const int E = in_sizes[0] / 32;     // key0 is [E,32,1]
    const int N = in_sizes[4] / 32;     // query0 is [N,32,1]
    const int npairs = E * NHEADS;      // 4,000,000 (multiple of 16)

    float* out0 = (float*)d_out;                 // [N,32,1] flat
    float* out1 = out0 + (size_t)N * 32;         // [N,32,3] flat

    // Workspace: logits/ex [E*H] | mmax [N*H] | ssum [N*H]  (~19.2 MB)
    char*  ws     = (char*)d_ws;
    float* logits = (float*)ws;
    float* mmax   = (float*)(ws + (size_t)npairs * sizeof(float));
    float* ssum   = mmax + (size_t)N * NHEADS;

    const long long out_n = (long long)N * 128;
    const long long nh    = (long long)N * NHEADS;
    {
        long long span = out_n;  // out_n > nh
        se3_init_kernel<<<(unsigned)((span + 255) / 256), 256, 0, stream>>>(
            out0, out_n, mmax, ssum, nh);
    }

    const int tiles   = (npairs + 15) / 16;
    const int blocksA = (tiles + 7) / 8;          // 8 waves (tiles) per 256-thread block
    se3_logits_wmma_kernel<<<blocksA, 256, 0, stream>>>(
        key0, key1, query0, query1, dst, logits, mmax, npairs);

    se3_exp_kernel<<<(npairs + 255) / 256, 256, 0, stream>>>(
        logits, dst, mmax, ssum, npairs);

    const long long total = (long long)E * 128;
    se3_agg_kernel<<<(unsigned)((total + 255) / 256), 256, 0, stream>>>(
        value0, value1, dst, logits, ssum, out0, out1, total);
}